// LocalizedTokenAggregation_8126078124233
// MI455X (gfx1250) — compile-verified
//
#include <hip/hip_runtime.h>
#include <cstddef>

#define L_DIM 512
#define B_DIM 128
#define D_DIM 512
#define C_DIM 32
#define TOPK  8

typedef __attribute__((ext_vector_type(2))) float v2f;
typedef __attribute__((ext_vector_type(8))) float v8f;

// ---------------------------------------------------------------------------
// Kernel 1: per-(l,b) row of C=32 concepts (one wave32 per row, lane = c).
// masked = tce>0 ? tce : -inf ; val_min = 8th largest (ties -> lane order,
// ranks are a permutation so exactly one lane has rank 7);
// keep = masked >= val_min  (IEEE: -inf >= -inf true -> all-kept when <8 pos)
// w = keep ? exp(masked) : 0   (exp(-inf)=0) ; pseudo_label = (w > 0)
// ---------------------------------------------------------------------------
__global__ void __launch_bounds__(256)
lta_topk_exp_kernel(const float* __restrict__ tce,
                    float* __restrict__ w,
                    float* __restrict__ pl)
{
    const int wave = threadIdx.x >> 5;
    const int lane = threadIdx.x & 31;
    const int row  = blockIdx.x * 8 + wave;          // row in [0, L*B)
    const float NEGINF = -__builtin_inff();

    float v = tce[(size_t)row * C_DIM + lane];
    v = (v > 0.0f) ? v : NEGINF;

    // rank of this lane's value (descending, ties broken by lane index)
    int rank = 0;
    #pragma unroll
    for (int j = 0; j < 32; ++j) {
        float vj = __shfl(v, j, 32);
        rank += ((vj > v) || ((vj == v) && (j < lane))) ? 1 : 0;
    }

    // broadcast the value whose rank == TOPK-1 (exactly one lane)
    float cand = (rank == (TOPK - 1)) ? v : NEGINF;
    #pragma unroll
    for (int off = 16; off > 0; off >>= 1)
        cand = fmaxf(cand, __shfl_xor(cand, off, 32));
    const float val_min = cand;

    const bool keep = (v >= val_min);
    const float wv  = keep ? __expf(v) : 0.0f;       // exp(-inf) == 0

    w [(size_t)row * C_DIM + lane] = wv;
    pl[(size_t)row * C_DIM + lane] = (wv > 0.0f) ? 1.0f : 0.0f;
}

// ---------------------------------------------------------------------------
// Kernel 2: colsum over L per (b,c); inv = sum>0 ? 1/sum : 0
// (sum==0 reproduces reference's NaN->0 softmax cleanup)
// ---------------------------------------------------------------------------
__global__ void __launch_bounds__(256)
lta_colsum_inv_kernel(const float* __restrict__ w, float* __restrict__ inv)
{
    const int t = blockIdx.x * 256 + threadIdx.x;    // 0 .. B*C-1
    float s = 0.0f;
    #pragma unroll 4
    for (int l = 0; l < L_DIM; ++l)
        s += w[(size_t)l * (B_DIM * C_DIM) + t];
    inv[t] = (s > 0.0f) ? (1.0f / s) : 0.0f;
}

// ---------------------------------------------------------------------------
// Kernel 3: merge_val[c,b,d] = inv[b,c] * sum_l w[l,b,c] * x[l,b,d]
// Per wave: M=32 (c), N=32 (d strip), K=L=512 via V_WMMA_F32_16X16X4_F32.
// A (16x4 f32): lane m<16 holds {K=k,K=k+1} of row M=m; lanes 16-31 {k+2,k+3}.
// B (4x16 f32): lane m<16 holds {K=k,K=k+1} of col N=m; lanes 16-31 {k+2,k+3}.
// C/D (16x16 f32): lane<16 vgpr j = D[M=j][N=lane]; lane>=16 -> M=j+8.
// ---------------------------------------------------------------------------
__global__ void __launch_bounds__(256)
lta_merge_wmma_kernel(const float* __restrict__ x,
                      const float* __restrict__ w,
                      const float* __restrict__ inv,
                      float* __restrict__ out)
{
    const int lane  = threadIdx.x & 31;
    const int wave  = threadIdx.x >> 5;
    const int b     = blockIdx.y;
    const int dbase = blockIdx.x * 256 + wave * 32;
    const int m     = lane & 15;
    const int half  = lane >> 4;                     // 0: K rows k,k+1 ; 1: k+2,k+3

    v8f acc00 = {}; v8f acc01 = {}; v8f acc10 = {}; v8f acc11 = {};

    const float* wB = w + (size_t)b * C_DIM;         // stride between l rows: B*C
    const float* xB = x + (size_t)b * D_DIM;         // stride between l rows: B*D

    for (int k = 0; k < L_DIM; k += 4) {
        const int l0 = k + 2 * half;

        const float* wa = wB + (size_t)l0 * (B_DIM * C_DIM);
        v2f a0, a1;                                   // A: M tiles c=0..15, c=16..31
        a0.x = wa[m];
        a0.y = wa[(B_DIM * C_DIM) + m];
        a1.x = wa[16 + m];
        a1.y = wa[(B_DIM * C_DIM) + 16 + m];

        const float* xb = xB + (size_t)l0 * (B_DIM * D_DIM);
        v2f b0, b1;                                   // B: N tiles d, d+16
        b0.x = xb[dbase + m];
        b0.y = xb[(B_DIM * D_DIM) + dbase + m];
        b1.x = xb[dbase + 16 + m];
        b1.y = xb[(B_DIM * D_DIM) + dbase + 16 + m];

        acc00 = __builtin_amdgcn_wmma_f32_16x16x4_f32(false, a0, false, b0,
                                                      (short)0, acc00, false, false);
        acc01 = __builtin_amdgcn_wmma_f32_16x16x4_f32(false, a0, false, b1,
                                                      (short)0, acc01, false, false);
        acc10 = __builtin_amdgcn_wmma_f32_16x16x4_f32(false, a1, false, b0,
                                                      (short)0, acc10, false, false);
        acc11 = __builtin_amdgcn_wmma_f32_16x16x4_f32(false, a1, false, b1,
                                                      (short)0, acc11, false, false);
    }

    // Epilogue: scale rows by softmax normalization and store merge_val (C,B,D)
    const float* invB = inv + (size_t)b * C_DIM;
    float* outB = out + (size_t)b * D_DIM;
    #pragma unroll
    for (int j = 0; j < 8; ++j) {
        const int c0 = half * 8 + j;                  // M tile 0
        const int c1 = 16 + half * 8 + j;             // M tile 1
        const float s0 = invB[c0];
        const float s1 = invB[c1];
        outB[(size_t)c0 * (B_DIM * D_DIM) + dbase + m]      = acc00[j] * s0;
        outB[(size_t)c0 * (B_DIM * D_DIM) + dbase + 16 + m] = acc01[j] * s0;
        outB[(size_t)c1 * (B_DIM * D_DIM) + dbase + m]      = acc10[j] * s1;
        outB[(size_t)c1 * (B_DIM * D_DIM) + dbase + 16 + m] = acc11[j] * s1;
    }
}

// ---------------------------------------------------------------------------
extern "C" void kernel_launch(void* const* d_in, const int* in_sizes, int n_in,
                              void* d_out, int out_size, void* d_ws, size_t ws_size,
                              hipStream_t stream)
{
    (void)in_sizes; (void)n_in; (void)out_size; (void)ws_size;

    const float* x   = (const float*)d_in[0];        // (L,B,D) f32
    const float* tce = (const float*)d_in[1];        // (L,B,C) f32

    float* out_merge = (float*)d_out;                              // (C,B,D)
    float* out_pl    = (float*)d_out + (size_t)C_DIM * B_DIM * D_DIM; // (L,B,C)

    float* w_buf   = (float*)d_ws;                                 // L*B*C floats
    float* inv_buf = w_buf + (size_t)L_DIM * B_DIM * C_DIM;        // B*C floats

    // 1) top-8 mask + exp weights + pseudo_label (1 wave per (l,b) row)
    lta_topk_exp_kernel<<<(L_DIM * B_DIM) / 8, 256, 0, stream>>>(tce, w_buf, out_pl);

    // 2) softmax denominators over L
    lta_colsum_inv_kernel<<<(B_DIM * C_DIM) / 256, 256, 0, stream>>>(w_buf, inv_buf);

    // 3) fp32 WMMA batched GEMM with fused normalization
    dim3 g3(D_DIM / 256, B_DIM);
    lta_merge_wmma_kernel<<<g3, 256, 0, stream>>>(x, w_buf, inv_buf, out_merge);
}